// MoE_11940009083455
// MI455X (gfx1250) — compile-verified
//
#include <hip/hip_runtime.h>

// GMM image renderer for MI455X (gfx1250, wave32).
// term1/term2 (affine in pixel coords) evaluated with V_WMMA_F32_16X16X4_F32:
//   D(16 gaussians x 16 pixels) = A(16x4 coeffs) x B(4x16 pixel basis)
// All mixture constants built once per wave from the 10.7KB param table.

typedef __attribute__((ext_vector_type(2))) float v2f;
typedef __attribute__((ext_vector_type(8))) float v8f;

#define GK 16        // gaussians per channel
#define A_CH 24      // B*CH = 8*3
#define H_DIM 256
#define N_PIX 65536  // 256*256
#define TILES 4096   // N_PIX / 16

__global__ __launch_bounds__(256) void gmm_render_kernel(
    const float* __restrict__ params, float* __restrict__ out) {
  const int lane = threadIdx.x & 31;
  const int g    = lane & 15;   // gaussian index (A rows; pixel index for B/D)
  const int hi   = lane >> 4;   // which half of the wave (selects K 0,1 vs 2,3)
  const int a    = blockIdx.y;  // channel 0..23

  // ---- per-wave setup: build per-gaussian constants from params[a] ----
  const float* pa = params + a * (7 * GK);
  const float mux = pa[g];
  const float muy = pa[GK + g];
  const float z   = pa[2 * GK + g];
  const float l00 = pa[3 * GK + 4 * g + 0];
  const float l10 = pa[3 * GK + 4 * g + 2];
  const float l11 = pa[3 * GK + 4 * g + 3];

  // sigma = L L^T row-sums (srow), affine constants
  const float srx = l00 * l00 + l00 * l10;
  const float sry = l00 * l10 + l10 * l10 + l11 * l11;
  const float c1c = -(srx * mux + sry * muy);
  const float c2c = -(mux + muy);

  // softmax over the 16 gaussians (16-lane butterfly; masks stay in-group)
  float mx = z;
  mx = fmaxf(mx, __shfl_xor(mx, 1, 32));
  mx = fmaxf(mx, __shfl_xor(mx, 2, 32));
  mx = fmaxf(mx, __shfl_xor(mx, 4, 32));
  mx = fmaxf(mx, __shfl_xor(mx, 8, 32));
  float ez = __expf(z - mx);
  float ssum = ez;
  ssum += __shfl_xor(ssum, 1, 32);
  ssum += __shfl_xor(ssum, 2, 32);
  ssum += __shfl_xor(ssum, 4, 32);
  ssum += __shfl_xor(ssum, 8, 32);
  const float w = ez / ssum;

  // Per-lane weights matching the D layout: VGPR r holds gaussian r + 8*hi.
  float wv[8];
#pragma unroll
  for (int r = 0; r < 8; ++r) wv[r] = __shfl(w, r + 8 * hi, 16);

  // A matrices (16x4 f32 layout: lanes 0-15 -> K0,K1 ; lanes 16-31 -> K2,K3)
  v2f a1, a2;
  a1.x = hi ? c1c : srx;
  a1.y = hi ? 0.0f : sry;
  a2.x = hi ? c2c : 1.0f;
  a2.y = hi ? 0.0f : 1.0f;

  const v8f czero = {};

  const int waves_per_blk = blockDim.x >> 5;
  const int wave   = blockIdx.x * waves_per_blk + (threadIdx.x >> 5);
  const int nwaves = gridDim.x * waves_per_blk;

  // ---- pixel-tile loop: 16 pixels per wave-iteration ----
  for (int tile = wave; tile < TILES; tile += nwaves) {
    const int pn = (tile << 4) + g;  // pixel index within channel
    const float gx = (float)(pn >> 8) * (1.0f / 255.0f);
    const float gy = (float)(pn & 255) * (1.0f / 255.0f);

    // B matrix (4x16 f32): lanes 0-15 rows K0,K1 = (gx,gy); 16-31 K2,K3 = (1,0)
    v2f b;
    b.x = hi ? 1.0f : gx;
    b.y = hi ? 0.0f : gy;

    v8f t1 = __builtin_amdgcn_wmma_f32_16x16x4_f32(
        false, a1, false, b, (short)0, czero, false, false);
    v8f t2 = __builtin_amdgcn_wmma_f32_16x16x4_f32(
        false, a2, false, b, (short)0, czero, false, false);

    float se = 0.0f, swe = 0.0f;
#pragma unroll
    for (int r = 0; r < 8; ++r) {
      const float e = __expf(-0.5f * t1[r] * t2[r]);
      se += e;
      swe = fmaf(wv[r], e, swe);
    }
    // combine the two gaussian halves (lane l <-> l+16 hold the same pixel)
    se  += __shfl_xor(se, 16, 32);
    swe += __shfl_xor(swe, 16, 32);

    float y = swe / fmaxf(se, 1e-8f);
    y = fminf(fmaxf(y, 0.0f), 1.0f);
    if (hi == 0) out[a * N_PIX + pn] = y;  // coalesced 16-lane store
  }
}

extern "C" void kernel_launch(void* const* d_in, const int* in_sizes, int n_in,
                              void* d_out, int out_size, void* d_ws, size_t ws_size,
                              hipStream_t stream) {
  (void)in_sizes; (void)n_in; (void)d_ws; (void)ws_size; (void)out_size;
  const float* params = (const float*)d_in[0];  // (8, 3, 112) fp32
  float* out = (float*)d_out;                   // (8, 3, 256, 256) fp32
  dim3 grid(64, A_CH);
  gmm_render_kernel<<<grid, 256, 0, stream>>>(params, out);
}